// _DeepECT_Module_28965259444796
// MI455X (gfx1250) — compile-verified
//
#include <hip/hip_runtime.h>

typedef __attribute__((ext_vector_type(2))) float v2f;
typedef __attribute__((ext_vector_type(8))) float v8f;

#define KCENT 256
#define DDIM  64

// ---------------------------------------------------------------------------
// Kernel 0: zero the per-leaf stats region of d_out (it is atomically
// accumulated into) and precompute ||c||^2 for all K centers into d_ws.
// ---------------------------------------------------------------------------
__global__ void prep_kernel(const float* __restrict__ centers,
                            float* __restrict__ stats,   // d_out + 2N, len nStats
                            float* __restrict__ csq,     // d_ws, len K
                            int nStats) {
  int tid = blockIdx.x * blockDim.x + threadIdx.x;
  if (tid < nStats) stats[tid] = 0.0f;
  if (tid < KCENT) {
    float s = 0.0f;
#pragma unroll
    for (int d = 0; d < DDIM; ++d) {
      float v = centers[tid * DDIM + d];
      s += v * v;
    }
    csq[tid] = s;
  }
}

// ---------------------------------------------------------------------------
// Kernel 1: per-wave 16-row x 256-center distance tile via
// V_WMMA_F32_16X16X4_F32 (exact f32, matches the f32 reference math).
//
// A layout (32-bit A, 16x4): lane L<16 holds A[M=L][K=4k+{0,1}] in .x/.y;
// lane L+16 holds A[M=L][K=4k+{2,3}].  B (4x16) mirrors the per-lane split;
// C/D: VGPR j = row {j | j+8}, lane%16 = column.
//
// Two center tiles are processed per loop iteration to give the matrix pipe
// two independent accumulation chains (the D->C dependency otherwise
// serializes back-to-back WMMAs from one wave).
//
// Squared distance is tracked through the argmin (sqrt is monotonic); the
// clamp max(d2,0) is applied before comparison to preserve reference tie
// semantics, and a single sqrt per row is taken at the end.
// ---------------------------------------------------------------------------
__global__ __launch_bounds__(128)
void dist_argmin_kernel(const float* __restrict__ X,
                        const float* __restrict__ C,
                        const float* __restrict__ csq,
                        float* __restrict__ assignOut,
                        float* __restrict__ mindOut,
                        float* __restrict__ counts,
                        float* __restrict__ sumsq) {
  __shared__ float sC[KCENT * DDIM];   // 64 KB of 320 KB/WGP
  __shared__ float sCsq[KCENT];        // ||c||^2, indexed per tile from LDS

  // Stage centers + ||c||^2 into LDS once per block (centers stay L2-resident).
  {
    const float4* src = (const float4*)C;
    float4* dst = (float4*)sC;
    for (int i = threadIdx.x; i < (KCENT * DDIM) / 4; i += blockDim.x)
      dst[i] = src[i];
    for (int i = threadIdx.x; i < KCENT; i += blockDim.x)
      sCsq[i] = csq[i];
  }
  __syncthreads();

  const int lane = threadIdx.x & 31;
  const int wave = threadIdx.x >> 5;
  const int half = lane >> 4;     // which K-pair / which row-half
  const int l16  = lane & 15;
  const long rowBase = (long)blockIdx.x * 64 + wave * 16;

  // Preload this wave's 16x64 A tile as 16 float2 fragments per lane.
  v2f af[16];
  float part = 0.0f;
#pragma unroll
  for (int kk = 0; kk < 16; ++kk) {
    af[kk] = *(const v2f*)&X[(rowBase + l16) * DDIM + kk * 4 + 2 * half];
    part += af[kk].x * af[kk].x + af[kk].y * af[kk].y;
  }
  // lane L and L^16 together hold row (L%16): combine for full ||x||^2.
  float xsq = part + __shfl_xor(part, 16, 32);   // lane L holds xsq[row L%16]

  // Hoisted: ||x||^2 for the row each C-slot j holds (loop-invariant).
  float xr[8];
#pragma unroll
  for (int j = 0; j < 8; ++j) xr[j] = __shfl(xsq, j + 8 * half, 16);

  float bestd2[8];
  int   besti[8];
#pragma unroll
  for (int j = 0; j < 8; ++j) { bestd2[j] = 3.4e38f; besti[j] = 0; }

  const int bofs = l16 * DDIM + 2 * half;   // per-lane base into sC

#pragma unroll 1
  for (int tp = 0; tp < 8; ++tp) {          // 8 iterations x 2 center tiles
    const int t0 = 2 * tp, t1 = 2 * tp + 1;
    const float cs0 = sCsq[t0 * 16 + l16];  // ||c||^2 for this lane's columns
    const float cs1 = sCsq[t1 * 16 + l16];
    v8f acc0 = {};
    v8f acc1 = {};
#pragma unroll
    for (int kk = 0; kk < 16; ++kk) {       // D = 64 in k-steps of 4
      v2f b0 = *(const v2f*)&sC[t0 * 16 * DDIM + bofs + kk * 4];
      v2f b1 = *(const v2f*)&sC[t1 * 16 * DDIM + bofs + kk * 4];
      acc0 = __builtin_amdgcn_wmma_f32_16x16x4_f32(
          false, af[kk], false, b0, (short)0, acc0, false, false);
      acc1 = __builtin_amdgcn_wmma_f32_16x16x4_f32(
          false, af[kk], false, b1, (short)0, acc1, false, false);
    }
    // acc[j] = dot(row j+8*half, center t*16+l16)
#pragma unroll
    for (int j = 0; j < 8; ++j) {
      float d20 = fmaxf(fmaf(-2.0f, acc0[j], xr[j] + cs0), 0.0f);
      float d21 = fmaxf(fmaf(-2.0f, acc1[j], xr[j] + cs1), 0.0f);
      int c0 = t0 * 16 + l16;
      int c1 = t1 * 16 + l16;
      if (d20 < bestd2[j]) { bestd2[j] = d20; besti[j] = c0; }  // strict <:
      if (d21 < bestd2[j]) { bestd2[j] = d21; besti[j] = c1; }  // first-occurrence
    }
  }

  // Reduce (min, argmin) across the 16 columns held in each 16-lane half.
#pragma unroll
  for (int j = 0; j < 8; ++j) {
    float d2 = bestd2[j];
    int   i  = besti[j];
#pragma unroll
    for (int off = 8; off >= 1; off >>= 1) {
      float od = __shfl_xor(d2, off, 16);
      int   oi = __shfl_xor(i, off, 16);
      if (od < d2 || (od == d2 && oi < i)) { d2 = od; i = oi; }
    }
    if (l16 == 0) {                 // lane0 -> rows 0..7, lane16 -> rows 8..15
      long row = rowBase + j + 8 * half;
      float d = sqrtf(d2);          // single sqrt per row, after reduction
      assignOut[row] = (float)i;
      mindOut[row]   = d;
      atomicAdd(&counts[i], 1.0f);
      atomicAdd(&sumsq[i], d * d);  // matches reference min_dist**2 rounding
    }
  }
}

// ---------------------------------------------------------------------------
// Kernel 2: per-leaf coordinate sums. One thread per (sample, dim) f32 atomic.
// ---------------------------------------------------------------------------
__global__ void leaf_sums_kernel(const float* __restrict__ X,
                                 const float* __restrict__ assignF,
                                 float* __restrict__ sums,
                                 long total) {
  long idx = (long)blockIdx.x * blockDim.x + threadIdx.x;
  if (idx >= total) return;
  long row = idx >> 6;           // DDIM == 64
  int  d   = (int)(idx & 63);
  int  c   = (int)assignF[row];  // values <= 255 are exact in f32
  atomicAdd(&sums[c * DDIM + d], X[idx]);
}

// ---------------------------------------------------------------------------
extern "C" void kernel_launch(void* const* d_in, const int* in_sizes, int n_in,
                              void* d_out, int out_size, void* d_ws, size_t ws_size,
                              hipStream_t stream) {
  const float* X = (const float*)d_in[0];   // [N, 64]
  const float* C = (const float*)d_in[1];   // [256, 64]
  const long N = (long)in_sizes[0] / DDIM;  // 131072

  float* out       = (float*)d_out;
  float* assignOut = out;                       // [N]
  float* mindOut   = out + N;                   // [N]
  float* counts    = out + 2 * N;               // [K]
  float* sums      = counts + KCENT;            // [K, 64]
  float* sumsq     = sums + KCENT * DDIM;       // [K]
  float* csq       = (float*)d_ws;              // [K]

  const int nStats = KCENT + KCENT * DDIM + KCENT;   // counts + sums + sumsq
  prep_kernel<<<(nStats + 255) / 256, 256, 0, stream>>>(C, counts, csq, nStats);

  const int blocks = (int)(N / 64);                  // 64 rows per block (4 waves)
  dist_argmin_kernel<<<blocks, 128, 0, stream>>>(X, C, csq, assignOut, mindOut,
                                                 counts, sumsq);

  const long total = N * DDIM;
  leaf_sums_kernel<<<(int)((total + 255) / 256), 256, 0, stream>>>(X, assignOut,
                                                                   sums, total);
}